// RGCN_80470507258384
// MI455X (gfx1250) — compile-verified
//
#include <hip/hip_runtime.h>
#include <hip/hip_bf16.h>
#include <math.h>

#define NNODES 50000
#define NEDGES 1600000
#define NREL   50
#define NBASES 30
#define NH     16
#define NC     16

#define PADDED_E ((NEDGES + NREL * 16 + 15) & ~15)   // 1,600,800
#define NTILES_E (PADDED_E / 16)                      // 100,050
#define NTILES_N (NNODES / 16)                        // 3,125
#define NHTOT    (NNODES * NH)                        // 800,000
#define NTILES_X (NHTOT / 16)                         // 50,000

typedef __attribute__((ext_vector_type(2))) float v2f;
typedef __attribute__((ext_vector_type(8))) float v8f;

static __device__ __forceinline__ v8f wmma_f32_16x16x4(v2f a, v2f b, v8f c) {
  // D = A(16x4, f32) * B(4x16, f32) + C(16x16, f32)
  return __builtin_amdgcn_wmma_f32_16x16x4_f32(false, a, false, b, (short)0, c,
                                               false, false);
}

// ---------------------------------------------------------------- histogram
__global__ void k_count(const int* __restrict__ ei, const int* __restrict__ et,
                        unsigned* __restrict__ counts,
                        unsigned* __restrict__ relCnt) {
  int e = blockIdx.x * blockDim.x + threadIdx.x;
  if (e >= NEDGES) return;
  int d = ei[NEDGES + e];
  int r = et[e];
  atomicAdd(&counts[(size_t)d * NREL + r], 1u);
  atomicAdd(&relCnt[r], 1u);
}

// -------------------------------------------------- prefix (padded to 16)
__global__ void k_scan(const unsigned* __restrict__ relCnt,
                       unsigned* __restrict__ relCur) {
  if (threadIdx.x != 0 || blockIdx.x != 0) return;
  unsigned off = 0;
  for (int r = 0; r < NREL; ++r) {
    relCur[r] = off;
    off += (relCnt[r] + 15u) & ~15u;
  }
}

// ----------------------------------------------------------- bucket scatter
__global__ void k_scatter(const int* __restrict__ ei, const int* __restrict__ et,
                          const unsigned* __restrict__ counts,
                          unsigned* __restrict__ relCur,
                          int* __restrict__ sSrc, int* __restrict__ sDst,
                          int* __restrict__ sRel, float* __restrict__ sInv) {
  int e = blockIdx.x * blockDim.x + threadIdx.x;
  if (e >= NEDGES) return;
  int s = ei[e];
  int d = ei[NEDGES + e];
  int r = et[e];
  unsigned pos = atomicAdd(&relCur[r], 1u);
  unsigned c = counts[(size_t)d * NREL + r];
  sSrc[pos] = s;
  sDst[pos] = d;
  sRel[pos] = r;
  sInv[pos] = 1.0f / (float)(c > 0u ? c : 1u);
}

// ---- w1[r,x] = sum_b comp1[r,b] * basis1[b,x]  as WMMA GEMM:
//      M = 50 (pad 64, 4 r-tiles), K = 30 (pad 32, 8 chunks), N = 800,000.
#define APAD 34  // LDS row stride (even for b64 alignment, avoids bank conflicts)
__global__ void k_w1(const float* __restrict__ basis1,
                     const float* __restrict__ comp1, float* __restrict__ w1) {
  __shared__ float sA[64 * APAD];  // zero-padded comp1
  for (int i = threadIdx.x; i < 64 * 32; i += blockDim.x) {
    int r = i >> 5, k = i & 31;
    sA[r * APAD + k] = (r < NREL && k < NBASES) ? comp1[r * NBASES + k] : 0.f;
  }
  __syncthreads();
  int wid = (blockIdx.x * blockDim.x + threadIdx.x) >> 5;
  int lane = threadIdx.x & 31;
  if (wid >= NTILES_X) return;  // wave-uniform
  int x0 = wid * 16;
  int m = lane & 15;
  int khalf = (lane < 16) ? 0 : 2;  // f32 WMMA lane striping over K
  v8f z = {0.f, 0.f, 0.f, 0.f, 0.f, 0.f, 0.f, 0.f};
  v8f acc[4] = {z, z, z, z};
#pragma unroll
  for (int kc = 0; kc < 8; ++kc) {
    int k = kc * 4 + khalf;  // always even
    v2f b;
    b.x = (k < NBASES) ? basis1[(size_t)k * NHTOT + x0 + m] : 0.f;
    b.y = (k + 1 < NBASES) ? basis1[(size_t)(k + 1) * NHTOT + x0 + m] : 0.f;
#pragma unroll
    for (int rt = 0; rt < 4; ++rt) {
      const float* ap = &sA[(rt * 16 + m) * APAD + k];
      v2f a; a.x = ap[0]; a.y = ap[1];  // ds_load_b64 (zero rows kill pad K)
      acc[rt] = wmma_f32_16x16x4(a, b, acc[rt]);
    }
  }
#pragma unroll
  for (int rt = 0; rt < 4; ++rt) {
#pragma unroll
    for (int v = 0; v < 8; ++v) {
      int row = rt * 16 + v + ((lane < 16) ? 0 : 8);
      if (row < NREL) w1[(size_t)row * NHTOT + x0 + m] = acc[rt][v];
    }
  }
}

// ---------------------------------------- w2[r,i,o] = sum_b comp2[r,b]*basis2
__global__ void k_w2(const float* __restrict__ basis2,
                     const float* __restrict__ comp2, float* __restrict__ w2) {
  int x = blockIdx.x * blockDim.x + threadIdx.x;
  if (x >= NREL * NH * NC) return;
  int r = x >> 8;
  int xx = x & 255;
  float acc = 0.f;
#pragma unroll
  for (int b = 0; b < NBASES; ++b)
    acc = fmaf(comp2[r * NBASES + b], basis2[b * (NH * NC) + xx], acc);
  w2[x] = acc;
}

// ------------------------------------------------------- h := root1 + bias1
__global__ void k_init1(const float* __restrict__ root1,
                        const float* __restrict__ bias1,
                        float* __restrict__ hbuf) {
  int x = blockIdx.x * blockDim.x + threadIdx.x;
  if (x >= NHTOT) return;
  hbuf[x] = root1[x] + bias1[x & (NH - 1)];
}

// -------------- layer-1 gather/scatter: h[dst] += inv * w1[r,src]
//                4 threads per edge, float4 gather + 4 f32 atomics each.
__global__ void k_edge1(const int* __restrict__ sSrc, const int* __restrict__ sDst,
                        const int* __restrict__ sRel,
                        const float* __restrict__ sInv,
                        const float* __restrict__ w1, float* __restrict__ hbuf) {
  int t = blockIdx.x * blockDim.x + threadIdx.x;
  int e = t >> 2;
  int q = (t & 3) * 4;
  if (e >= PADDED_E) return;
  int d = sDst[e];
  if (d < 0) return;  // padding slot
  int s = sSrc[e];
  int r = sRel[e];
  float inv = sInv[e];
  const float4 v = *(const float4*)(w1 + (size_t)r * NHTOT + (size_t)s * NH + q);
  float* o = hbuf + (size_t)d * NH + q;
  atomicAdd(o + 0, v.x * inv);
  atomicAdd(o + 1, v.y * inv);
  atomicAdd(o + 2, v.z * inv);
  atomicAdd(o + 3, v.w * inv);
}

__global__ void k_relu(float* __restrict__ hbuf) {
  int x = blockIdx.x * blockDim.x + threadIdx.x;
  if (x >= NHTOT) return;
  hbuf[x] = fmaxf(hbuf[x], 0.f);
}

// ------------------------ out := h @ root2 + bias2   (WMMA f32 16x16x4, K=16)
__global__ void k_init_out(const float* __restrict__ hbuf,
                           const float* __restrict__ root2,
                           const float* __restrict__ bias2,
                           float* __restrict__ out) {
  int wid = (blockIdx.x * blockDim.x + threadIdx.x) >> 5;
  int lane = threadIdx.x & 31;
  if (wid >= NTILES_N) return;  // wave-uniform
  int n0 = wid * 16;
  int m = lane & 15;                    // A row / B col for this lane
  int khalf = (lane < 16) ? 0 : 2;
  const float* hrow = hbuf + (size_t)(n0 + m) * NH;
  v8f c = {0.f, 0.f, 0.f, 0.f, 0.f, 0.f, 0.f, 0.f};
#pragma unroll
  for (int kc = 0; kc < 4; ++kc) {
    int k = kc * 4 + khalf;
    v2f a; a.x = hrow[k];               a.y = hrow[k + 1];
    v2f b; b.x = root2[k * NC + m];     b.y = root2[(k + 1) * NC + m];
    c = wmma_f32_16x16x4(a, b, c);
  }
  float bi = bias2[m];
#pragma unroll
  for (int v = 0; v < 8; ++v) {
    int row = v + ((lane < 16) ? 0 : 8);
    out[(size_t)(n0 + row) * NC + m] = c[v] + bi;
  }
}

// -------- layer-2: per 16-edge tile (one relation): D = h[src] @ w2[r],
//          then out[dst] += inv * D   (WMMA f32, atomic scatter)
__global__ void k_edge2(const int* __restrict__ sSrc, const int* __restrict__ sDst,
                        const int* __restrict__ sRel,
                        const float* __restrict__ sInv,
                        const float* __restrict__ hbuf,
                        const float* __restrict__ w2, float* __restrict__ out) {
  int wid = (blockIdx.x * blockDim.x + threadIdx.x) >> 5;
  int lane = threadIdx.x & 31;
  if (wid >= NTILES_E) return;  // wave-uniform
  int p0 = wid * 16;
  int m = lane & 15;
  int khalf = (lane < 16) ? 0 : 2;
  int rel = sRel[p0];  // whole tile is one relation (segments are 16-padded)
  const float* w2r = w2 + (size_t)rel * (NH * NC);
  int srcM = sSrc[p0 + m];
  const float* hrow = hbuf + (size_t)srcM * NH;
  v8f c = {0.f, 0.f, 0.f, 0.f, 0.f, 0.f, 0.f, 0.f};
#pragma unroll
  for (int kc = 0; kc < 4; ++kc) {
    int k = kc * 4 + khalf;
    v2f a; a.x = hrow[k];            a.y = hrow[k + 1];
    v2f b; b.x = w2r[k * NC + m];    b.y = w2r[(k + 1) * NC + m];
    c = wmma_f32_16x16x4(a, b, c);
  }
#pragma unroll
  for (int v = 0; v < 8; ++v) {
    int row = v + ((lane < 16) ? 0 : 8);
    int d = sDst[p0 + row];
    if (d >= 0) {
      float inv = sInv[p0 + row];
      atomicAdd(&out[(size_t)d * NC + m], c[v] * inv);
    }
  }
}

// ----------------------------------------------- row-wise log_softmax (16)
__global__ void k_logsoftmax(float* __restrict__ out) {
  int t = blockIdx.x * blockDim.x + threadIdx.x;
  int row = t >> 4;
  int col = t & 15;
  if (row >= NNODES) return;
  float v = out[(size_t)row * NC + col];
  float mx = v;
#pragma unroll
  for (int off = 1; off < 16; off <<= 1) mx = fmaxf(mx, __shfl_xor(mx, off, 16));
  float ex = expf(v - mx);
  float s = ex;
#pragma unroll
  for (int off = 1; off < 16; off <<= 1) s += __shfl_xor(s, off, 16);
  out[(size_t)row * NC + col] = v - mx - logf(s);
}

// ======================================================================
extern "C" void kernel_launch(void* const* d_in, const int* in_sizes, int n_in,
                              void* d_out, int out_size, void* d_ws,
                              size_t ws_size, hipStream_t stream) {
  (void)in_sizes; (void)n_in; (void)out_size; (void)ws_size;

  const int*   ei     = (const int*)d_in[0];   // [2,E]
  const int*   et     = (const int*)d_in[1];   // [E]
  const float* basis1 = (const float*)d_in[2]; // [B,N,H]
  const float* comp1  = (const float*)d_in[3]; // [R,B]
  const float* root1  = (const float*)d_in[4]; // [N,H]
  const float* bias1  = (const float*)d_in[5]; // [H]
  const float* basis2 = (const float*)d_in[6]; // [B,H,C]
  const float* comp2  = (const float*)d_in[7]; // [R,B]
  const float* root2  = (const float*)d_in[8]; // [H,C]
  const float* bias2  = (const float*)d_in[9]; // [C]
  float* out = (float*)d_out;                  // [N,C]

  // ---- workspace layout (256B aligned) ----
  char* ws = (char*)d_ws;
  size_t off = 0;
  auto alloc = [&](size_t bytes) {
    char* p = ws + off;
    off += (bytes + 255) & ~(size_t)255;
    return p;
  };
  unsigned* counts = (unsigned*)alloc((size_t)NNODES * NREL * 4); // 10 MB
  unsigned* relCnt = (unsigned*)alloc(64 * 4);
  unsigned* relCur = (unsigned*)alloc(64 * 4);
  int*   sSrc = (int*)alloc((size_t)PADDED_E * 4);
  int*   sDst = (int*)alloc((size_t)PADDED_E * 4);
  int*   sRel = (int*)alloc((size_t)PADDED_E * 4);
  float* sInv = (float*)alloc((size_t)PADDED_E * 4);
  float* w2   = (float*)alloc((size_t)NREL * NH * NC * 4);        // 50 KB
  float* hbuf = (float*)alloc((size_t)NHTOT * 4);                 // 3.2 MB
  float* w1   = (float*)alloc((size_t)NREL * NHTOT * 4);          // 160 MB

  // ---- clear / sentinel init ----
  hipMemsetAsync(counts, 0, (size_t)NNODES * NREL * 4, stream);
  hipMemsetAsync(relCnt, 0, 64 * 4, stream);
  hipMemsetAsync(sDst, 0xFF, (size_t)PADDED_E * 4, stream);  // dst = -1 pads
  hipMemsetAsync(sInv, 0, (size_t)PADDED_E * 4, stream);
  hipMemsetAsync(sSrc, 0, (size_t)PADDED_E * 4, stream);
  hipMemsetAsync(sRel, 0, (size_t)PADDED_E * 4, stream);

  const int BT = 256;
  // 1) per-(dst,rel) histogram + relation histogram
  k_count<<<(NEDGES + BT - 1) / BT, BT, 0, stream>>>(ei, et, counts, relCnt);
  // 2) 16-padded prefix offsets
  k_scan<<<1, 1, 0, stream>>>(relCnt, relCur);
  // 3) bucket edges by relation; record inv = 1/count(dst,rel)
  k_scatter<<<(NEDGES + BT - 1) / BT, BT, 0, stream>>>(ei, et, counts, relCur,
                                                       sSrc, sDst, sRel, sInv);
  // 4) dense weights (w1 via WMMA GEMM: 8 waves/block, 50k x-tiles)
  k_w1<<<(NTILES_X + 7) / 8, BT, 0, stream>>>(basis1, comp1, w1);
  k_w2<<<(NREL * NH * NC + BT - 1) / BT, BT, 0, stream>>>(basis2, comp2, w2);
  // 5) layer 1
  k_init1<<<(NHTOT + BT - 1) / BT, BT, 0, stream>>>(root1, bias1, hbuf);
  k_edge1<<<((size_t)PADDED_E * 4 + BT - 1) / BT, BT, 0, stream>>>(
      sSrc, sDst, sRel, sInv, w1, hbuf);
  k_relu<<<(NHTOT + BT - 1) / BT, BT, 0, stream>>>(hbuf);
  // 6) layer 2: out = h@root2 + bias2, then WMMA edge tiles accumulate
  k_init_out<<<(NTILES_N * 32 + BT - 1) / BT, BT, 0, stream>>>(hbuf, root2,
                                                               bias2, out);
  k_edge2<<<(NTILES_E * 32 + BT - 1) / BT, BT, 0, stream>>>(
      sSrc, sDst, sRel, sInv, hbuf, w2, out);
  // 7) log_softmax in place
  k_logsoftmax<<<(NNODES * NC + BT - 1) / BT, BT, 0, stream>>>(out);
}